// K_Rectify_38216619000515
// MI455X (gfx1250) — compile-verified
//
#include <hip/hip_runtime.h>
#include <stdint.h>

// Problem constants (match reference)
#define B_       128
#define NTOT_    129
#define N_       128
#define GS_      16
#define C_       384
#define EPS_     0.05f
#define RMS_EPS_ 1e-6f

#define WPB 2            // waves per block
#define TPB (WPB * 32)

typedef unsigned int u32x4 __attribute__((ext_vector_type(4)));
typedef unsigned int u32x8 __attribute__((ext_vector_type(8)));

__global__ __launch_bounds__(TPB)
void K_rectify_tdm_gather(const float* __restrict__ f,
                          const float* __restrict__ distance,
                          const float* __restrict__ rf,
                          const float* __restrict__ knorm_w,
                          const int*   __restrict__ idx,
                          float* __restrict__ out)
{
    // Per-wave staging: 16 gathered neighbor rows of 384 floats (TDM dest).
    __shared__ __align__(16) float stage[WPB][GS_][C_];

    const int lane = threadIdx.x & 31;
    const int wv   = threadIdx.x >> 5;
    const int wid  = blockIdx.x * WPB + wv;      // point id in [0, B*N)
    const int b    = wid >> 7;                   // wid / N
    const int n    = wid & (N_ - 1);             // wid % N

    const float* xrow  = f   + ((size_t)b * NTOT_ + 1 + n) * C_;   // x[b,n,:]
    const float* rfrow = rf  + (size_t)(1 + n) * C_;               // rf[1+n,:]
    float*       orow  = out + ((size_t)b * NTOT_ + 1 + n) * C_;

    // ---------------- inverse-distance weights (per-wave uniform 16) -------
    // lanes 16..31 mirror lanes 0..15 so xor-reduce over 16 stays coherent
    const int    g16   = lane & 15;
    const size_t pbase = ((size_t)b * N_ + n) * GS_;

    float d  = distance[pbase + g16];
    float r  = 1.0f / (d + EPS_);
    float rs = r;
    rs += __shfl_xor(rs, 1, 32);
    rs += __shfl_xor(rs, 2, 32);
    rs += __shfl_xor(rs, 4, 32);
    rs += __shfl_xor(rs, 8, 32);
    const float w = r / rs;                      // lane g16 holds w[g16]

    // ---------------- neighbor row index in f (fits 16 bits: < 16512) ------
    unsigned ri   = (unsigned)idx[pbase + g16];          // row in x, [0, B*N)
    unsigned frow = ri + (ri >> 7) + 1u;                 // (ri/128)*129 + 1 + ri%128

    // pack the 16 row indices into 8 dwords (16-bit pairs) for D# groups 2/3
    unsigned pk[8];
#pragma unroll
    for (int j = 0; j < 8; ++j) {
        unsigned lo = (unsigned)__shfl((int)frow, 2 * j,     32);
        unsigned hi = (unsigned)__shfl((int)frow, 2 * j + 1, 32);
        pk[j] = lo | (hi << 16);
    }

    // ---------------- Tensor DMA descriptor (gather mode) ------------------
    const uint64_t fa      = (uint64_t)(uintptr_t)f;
    const unsigned ldsaddr = (unsigned)(uintptr_t)&stage[wv][0][0];

    // group0: count=1 | gather_mode(bit31), 16-bit indices (bit30=0),
    //         lds_addr, global_addr[31:0], global_addr[56:32] | type=2(<<30)
    u32x4 g0 = { 0x80000001u,
                 ldsaddr,
                 (unsigned)fa,
                 ((unsigned)(fa >> 32) & 0x01FFFFFFu) | 0x80000000u };

    // group1 bit layout (32-bit words w0..w7):
    //  w0: data_size=2 (4B) at [17:16]; mask/flags zero
    //  w1: tensor_dim0[15:0] << 16            (tensor_dim0 = 384)
    //  w2: tensor_dim0[31:16]=0 | tensor_dim1[15:0] << 16  (tensor_dim1 = 16512 rows of f)
    //  w3: tensor_dim1[31:16]=0 | tile_dim0 << 16          (tile_dim0 = 384)
    //  w4: tile_dim1 = 16 (#valid gather indices), tile_dim2 = 0
    //  w5: tensor_dim0_stride[31:0] = 384
    //  w6,w7: stride hi / dim1 stride (ignored in gather mode) = 0
    u32x8 g1 = { 0x00020000u,
                 (384u & 0xFFFFu) << 16,
                 (16512u & 0xFFFFu) << 16,
                 (384u) << 16,
                 16u,
                 384u,
                 0u,
                 0u };

    u32x4 g2 = { pk[0], pk[1], pk[2], pk[3] };   // row_index_0..7
    u32x4 g3 = { pk[4], pk[5], pk[6], pk[7] };   // row_index_8..15

    // one TDM op gathers all 16 rows (24KB) into this wave's LDS stage
    asm volatile("tensor_load_to_lds %0, %1, %2, %3"
                 :: "s"(g0), "s"(g1), "s"(g2), "s"(g3)
                 : "memory");

    // ---------------- overlap TDM latency: center/rf/knorm + cls copy ------
    float4 xv[3], rv[3], kv[3];
#pragma unroll
    for (int k = 0; k < 3; ++k) {
        const int c = k * 128 + lane * 4;
        xv[k] = *(const float4*)(xrow  + c);
        rv[k] = *(const float4*)(rfrow + c);
        kv[k] = *(const float4*)(knorm_w + c);
    }
    if (n == 0) {
        const float* crow  = f   + (size_t)b * NTOT_ * C_;   // f[b,0,:]
        float*       corow = out + (size_t)b * NTOT_ * C_;   // out[b,0,:]
#pragma unroll
        for (int k = 0; k < 3; ++k) {
            const int c = k * 128 + lane * 4;
            *(float4*)(corow + c) = *(const float4*)(crow + c);
        }
    }

    // ---------------- wait for TDM, then weighted accumulate from LDS ------
    asm volatile("s_wait_tensorcnt 0x0" ::: "memory");

    float ax[3] = {0.f, 0.f, 0.f}, ay[3] = {0.f, 0.f, 0.f};
    float az[3] = {0.f, 0.f, 0.f}, aw[3] = {0.f, 0.f, 0.f};
#pragma unroll
    for (int g = 0; g < GS_; ++g) {
        const float wg = __shfl(w, g, 32);
#pragma unroll
        for (int k = 0; k < 3; ++k) {
            float4 v = *(const float4*)&stage[wv][g][k * 128 + lane * 4];
            ax[k] = fmaf(wg, v.x, ax[k]);
            ay[k] = fmaf(wg, v.y, ay[k]);
            az[k] = fmaf(wg, v.z, az[k]);
            aw[k] = fmaf(wg, v.w, aw[k]);
        }
    }

    // sf = sum_g w_g * neighbor - x ; accumulate sum of squares
    float sfx[3], sfy[3], sfz[3], sfw[3];
    float ss = 0.f;
#pragma unroll
    for (int k = 0; k < 3; ++k) {
        sfx[k] = ax[k] - xv[k].x;
        sfy[k] = ay[k] - xv[k].y;
        sfz[k] = az[k] - xv[k].z;
        sfw[k] = aw[k] - xv[k].w;
        ss = fmaf(sfx[k], sfx[k], ss);
        ss = fmaf(sfy[k], sfy[k], ss);
        ss = fmaf(sfz[k], sfz[k], ss);
        ss = fmaf(sfw[k], sfw[k], ss);
    }
    // wave-wide all-reduce (32 lanes x 12 cols = 384 cols exactly once)
    ss += __shfl_xor(ss, 1, 32);
    ss += __shfl_xor(ss, 2, 32);
    ss += __shfl_xor(ss, 4, 32);
    ss += __shfl_xor(ss, 8, 32);
    ss += __shfl_xor(ss, 16, 32);

    const float inv = 1.0f / sqrtf(ss * (1.0f / (float)C_) + RMS_EPS_);

    // out = rf + x + rmsnorm(sf) * knorm_w
#pragma unroll
    for (int k = 0; k < 3; ++k) {
        float4 o;
        o.x = rv[k].x + xv[k].x + sfx[k] * inv * kv[k].x;
        o.y = rv[k].y + xv[k].y + sfy[k] * inv * kv[k].y;
        o.z = rv[k].z + xv[k].z + sfz[k] * inv * kv[k].z;
        o.w = rv[k].w + xv[k].w + sfw[k] * inv * kv[k].w;
        const int c = k * 128 + lane * 4;
        *(float4*)(orow + c) = o;
    }
}

extern "C" void kernel_launch(void* const* d_in, const int* in_sizes, int n_in,
                              void* d_out, int out_size, void* d_ws, size_t ws_size,
                              hipStream_t stream)
{
    (void)in_sizes; (void)n_in; (void)out_size; (void)d_ws; (void)ws_size;
    const float* f        = (const float*)d_in[0];
    const float* distance = (const float*)d_in[1];
    const float* rf       = (const float*)d_in[2];
    const float* knorm_w  = (const float*)d_in[3];
    const int*   idx      = (const int*)d_in[4];
    float*       out      = (float*)d_out;

    const int total_points = B_ * N_;             // 16384 waves
    dim3 grid(total_points / WPB);
    dim3 block(TPB);
    K_rectify_tdm_gather<<<grid, block, 0, stream>>>(f, distance, rf, knorm_w, idx, out);
}